// TestMPNN_3_84507776516159
// MI455X (gfx1250) — compile-verified
//
#include <hip/hip_runtime.h>
#include <math.h>

// ---------------------------------------------------------------------------
// TestMPNN forward for MI455X (gfx1250, wave32).
// B=4, N=1024, L=3, K_top=64. fp32 throughout (matches reference numerics).
// The per-edge 3->32 affine uses V_WMMA_F32_16X16X4_F32 with a bias-ones
// column, so K=4 is fully utilized and the result is exact fp32.
// ---------------------------------------------------------------------------

typedef float v2f __attribute__((ext_vector_type(2)));
typedef float v8f __attribute__((ext_vector_type(8)));

#define NN   1024
#define BB   4
#define NL   3
#define KTOP 64

__device__ __forceinline__ float lrelu(float x) {
  // select form: v_cmp + v_cndmask + v_mul (avoids fmaxf NaN-quiet extra op)
  return x > 0.0f ? x : 0.01f * x;
}

__device__ __forceinline__ float waveReduceSum(float x) {
#pragma unroll
  for (int m = 16; m >= 1; m >>= 1) x += __shfl_xor(x, m, 32);
  return x;
}

// ---- power iteration: tmp[row] = E[row,:] . v[b,:] (one wave per row) ------
__global__ void matvec_kernel(const float* __restrict__ E,
                              const float* __restrict__ vin,
                              float* __restrict__ vout) {
  const int lane = threadIdx.x & 31;
  const int row  = blockIdx.x * (blockDim.x >> 5) + (threadIdx.x >> 5); // B*N rows
  const int b    = row >> 10;
  const float* Er = E + (size_t)row * NN;
  const float* vb = vin + (b << 10);
  float acc = 0.f;
#pragma unroll 4
  for (int j = lane; j < NN; j += 32) acc = fmaf(Er[j], vb[j], acc);
  acc = waveReduceSum(acc);
  if (lane == 0) vout[row] = acc;
}

// ---- per-batch L2 normalization with clip(1e-5) ----------------------------
__global__ void normalize_kernel(const float* __restrict__ tin,
                                 float* __restrict__ vout) {
  __shared__ float red[256];
  const int b = blockIdx.x, t = threadIdx.x;
  const float* tb = tin + b * NN;
  float s = 0.f;
  for (int j = t; j < NN; j += 256) { float x = tb[j]; s = fmaf(x, x, s); }
  red[t] = s;
  __syncthreads();
  for (int w = 128; w >= 1; w >>= 1) {
    if (t < w) red[t] += red[t + w];
    __syncthreads();
  }
  const float inv = 1.0f / fmaxf(sqrtf(red[0]), 1e-5f);
  for (int j = t; j < NN; j += 256) vout[b * NN + j] = tb[j] * inv;
}

// ---- message layer: agg[b,i] = mean_j( w2 . lrelu(A_ij @ Wb) + b2 ) --------
// One wave per row i. Per 32 j's: two 16-edge tiles, each needing 2 WMMAs
// (hidden 0..15 and 16..31). A[M=edge, K] = [v_i, v_j, E_ij, 1];
// B[K, N=hidden] = [W0; W1; W2; b1]. All exact fp32.
// Four independent accumulators break the epilogue FMA dependency chain
// (4 chains of 8 instead of 1 chain of 32) so dual-issue can interleave.
__global__ void msg_agg_kernel(const float* __restrict__ E,
                               const float* __restrict__ v,
                               const float* __restrict__ mw1,  // [L,3,32]
                               const float* __restrict__ mb1,  // [L,32]
                               const float* __restrict__ mw2,  // [L,32]
                               const float* __restrict__ mb2,  // [L]
                               float* __restrict__ agg, int l) {
  const int lane = threadIdx.x & 31;
  const int row  = blockIdx.x * (blockDim.x >> 5) + (threadIdx.x >> 5);
  const int b    = row >> 10;
  const int hs   = lane & 15;
  const bool lo16 = (lane < 16);

  const float* W  = mw1 + l * 96;   // W0=W[0..31], W1=W[32..63], W2=W[64..95]
  const float* B1 = mb1 + l * 32;
  const float* W2 = mw2 + l * 32;

  // B-matrix (4x16, K x N): VGPR0 = {K0 rows | K2 rows}, VGPR1 = {K1 | K3}.
  v2f Blo, Bhi;
  Blo.x = lo16 ? W[hs]           : W[64 + hs];       // K0=W0 / K2=W2
  Blo.y = lo16 ? W[32 + hs]      : B1[hs];           // K1=W1 / K3=b1
  Bhi.x = lo16 ? W[16 + hs]      : W[80 + hs];
  Bhi.y = lo16 ? W[48 + hs]      : B1[16 + hs];
  const float w2a = W2[hs], w2b = W2[16 + hs];

  const float  vi = v[row];
  const float* Er = E + (size_t)row * NN;
  const float* vb = v + (b << 10);

  float acc0 = 0.f, acc1 = 0.f, acc2 = 0.f, acc3 = 0.f;
  for (int j0 = 0; j0 < NN; j0 += 32) {
    const float e  = Er[j0 + lane];   // coalesced 128B
    const float vj = vb[j0 + lane];
    // shuffles hoisted out of lane-predicated selects (EXEC must stay full)
    const float eL  = __shfl(e,  hs, 32);        // E[j0 + hs]       (tile 0, K2 lanes)
    const float vjH = __shfl(vj, hs + 16, 32);   // v[j0 + 16 + hs]  (tile 1, K1 lanes)

    v2f A0, A1;  // A 16x4: lanes 0-15 -> K0,K1; lanes 16-31 -> K2,K3
    A0.x = lo16 ? vi : eL;   A0.y = lo16 ? vj  : 1.0f;
    A1.x = lo16 ? vi : e;    A1.y = lo16 ? vjH : 1.0f;

    const v8f z = {0.f, 0.f, 0.f, 0.f, 0.f, 0.f, 0.f, 0.f};
    v8f c0 = __builtin_amdgcn_wmma_f32_16x16x4_f32(false, A0, false, Blo, (short)0, z, false, false);
    v8f c1 = __builtin_amdgcn_wmma_f32_16x16x4_f32(false, A0, false, Bhi, (short)0, z, false, false);
    v8f c2 = __builtin_amdgcn_wmma_f32_16x16x4_f32(false, A1, false, Blo, (short)0, z, false, false);
    v8f c3 = __builtin_amdgcn_wmma_f32_16x16x4_f32(false, A1, false, Bhi, (short)0, z, false, false);

    // C layout: VGPR k <-> edge, lane <-> hidden. Sum everything into
    // per-lane scalars; the final wave reduction collapses hidden+edge sums.
#pragma unroll
    for (int k = 0; k < 8; ++k) {
      acc0 = fmaf(lrelu(c0[k]), w2a, acc0);
      acc1 = fmaf(lrelu(c1[k]), w2b, acc1);
      acc2 = fmaf(lrelu(c2[k]), w2a, acc2);
      acc3 = fmaf(lrelu(c3[k]), w2b, acc3);
    }
  }
  float acc = (acc0 + acc1) + (acc2 + acc3);
  acc = waveReduceSum(acc);
  if (lane == 0) agg[row] = acc * (1.0f / NN) + mb2[l];  // mean + b2
}

// ---- node update MLP (2->16->1) + gated residual ---------------------------
__global__ void update_kernel(float* __restrict__ v, const float* __restrict__ agg,
                              const float* __restrict__ w1,  // [L,2,16]
                              const float* __restrict__ b1,  // [L,16]
                              const float* __restrict__ w2,  // [L,16]
                              const float* __restrict__ b2,  // [L]
                              const float* __restrict__ logit_alpha, int l) {
  const int t = blockIdx.x * blockDim.x + threadIdx.x;  // B*N
  const float vi = v[t], ai = agg[t];
  const float* W1 = w1 + l * 32;
  const float* B1 = b1 + l * 16;
  const float* W2 = w2 + l * 16;
  float out = b2[l];
#pragma unroll
  for (int h = 0; h < 16; ++h) {
    const float hh = lrelu(vi * W1[h] + ai * W1[16 + h] + B1[h]);
    out = fmaf(hh, W2[h], out);
  }
  const float alpha = 0.85f / (1.0f + expf(-logit_alpha[l]));
  v[t] = alpha * vi + (1.0f - alpha) * out;
}

// ---- time embedding + tiny decoder (2->4->1) readout -----------------------
__global__ void readout_kernel(const float* __restrict__ v,
                               const float* __restrict__ time_arr,  // [B]
                               const float* __restrict__ log_scale,
                               const float* __restrict__ tw1,  // [2,4]
                               const float* __restrict__ tb1,  // [4]
                               const float* __restrict__ tw2,  // [4]
                               const float* __restrict__ tb2,  // [1]
                               const float* __restrict__ rc,   // [1]
                               float* __restrict__ sc) {
  const int t = blockIdx.x * blockDim.x + threadIdx.x;
  const int b = t >> 10;
  const float vi = v[t];
  const float te = tanhf(expf(log_scale[0]) * time_arr[b] * (1.0f / NN));
  float dec = tb2[0];
#pragma unroll
  for (int h = 0; h < 4; ++h) {
    const float hd = lrelu(vi * tw1[h] + te * tw1[4 + h] + tb1[h]);
    dec = fmaf(hd, tw2[h], dec);
  }
  sc[t] = rc[0] * (vi * te) + dec;
}

// ---- exact top-64 mask via rank counting (lower index wins ties) -----------
__global__ void topk_mask_kernel(float* __restrict__ sc) {
  __shared__ float a[NN];
  const int b = blockIdx.x, t = threadIdx.x;
  const float s  = sc[b * NN + t];
  const float av = fabsf(s);
  a[t] = av;
  __syncthreads();
  int rank = 0;
  for (int j = 0; j < NN; ++j) {
    const float aj = a[j];
    rank += (aj > av) || (aj == av && j < t);
  }
  sc[b * NN + t] = (rank < KTOP) ? s : 0.0f;
}

// ---- out[b,i,j] = s_i * s_j  (16 MB coalesced float4 stores) ---------------
__global__ void outer_kernel(const float* __restrict__ sc, float4* __restrict__ out) {
  const int t   = blockIdx.x * blockDim.x + threadIdx.x;  // B*N*N/4
  const int b   = t >> 18;            // N*N/4 = 2^18
  const int rem = t & ((1 << 18) - 1);
  const int i   = rem >> 8;           // N/4 = 256
  const int j4  = (rem & 255) << 2;
  const float si = sc[b * NN + i];
  const float* sj = sc + b * NN + j4;
  float4 o;
  o.x = si * sj[0]; o.y = si * sj[1]; o.z = si * sj[2]; o.w = si * sj[3];
  out[t] = o;
}

extern "C" void kernel_launch(void* const* d_in, const int* in_sizes, int n_in,
                              void* d_out, int out_size, void* d_ws, size_t ws_size,
                              hipStream_t stream) {
  const float* X           = (const float*)d_in[0];
  const float* E           = (const float*)d_in[1];
  const float* time_arr    = (const float*)d_in[2];
  const float* msg_w1      = (const float*)d_in[3];
  const float* msg_b1      = (const float*)d_in[4];
  const float* msg_w2      = (const float*)d_in[5];
  const float* msg_b2      = (const float*)d_in[6];
  const float* upd_w1      = (const float*)d_in[7];
  const float* upd_b1      = (const float*)d_in[8];
  const float* upd_w2      = (const float*)d_in[9];
  const float* upd_b2      = (const float*)d_in[10];
  const float* logit_alpha = (const float*)d_in[11];
  const float* log_scale   = (const float*)d_in[12];
  const float* tiny_w1     = (const float*)d_in[13];
  const float* tiny_b1     = (const float*)d_in[14];
  const float* tiny_w2     = (const float*)d_in[15];
  const float* tiny_b2     = (const float*)d_in[16];
  const float* readout_c   = (const float*)d_in[17];
  float* out = (float*)d_out;

  float* v   = (float*)d_ws;        // [B*N]
  float* tmp = v   + BB * NN;       // [B*N]
  float* agg = tmp + BB * NN;       // [B*N]
  float* sc  = agg + BB * NN;       // [B*N]

  const int rows     = BB * NN;     // 4096
  const int mvBlocks = rows / 8;    // 8 waves (256 thr) per block

  // ---- power iteration (3x): v = normalize(E @ v), clip 1e-5 ----
  matvec_kernel<<<mvBlocks, 256, 0, stream>>>(E, X, tmp);
  normalize_kernel<<<BB, 256, 0, stream>>>(tmp, v);
  for (int it = 1; it < 3; ++it) {
    matvec_kernel<<<mvBlocks, 256, 0, stream>>>(E, v, tmp);
    normalize_kernel<<<BB, 256, 0, stream>>>(tmp, v);
  }

  // ---- message-passing layers ----
  for (int l = 0; l < NL; ++l) {
    msg_agg_kernel<<<mvBlocks, 256, 0, stream>>>(E, v, msg_w1, msg_b1,
                                                 msg_w2, msg_b2, agg, l);
    update_kernel<<<rows / 256, 256, 0, stream>>>(v, agg, upd_w1, upd_b1,
                                                  upd_w2, upd_b2, logit_alpha, l);
  }

  // ---- readout + hard top-k + sparse outer product ----
  readout_kernel<<<rows / 256, 256, 0, stream>>>(v, time_arr, log_scale, tiny_w1,
                                                 tiny_b1, tiny_w2, tiny_b2,
                                                 readout_c, sc);
  topk_mask_kernel<<<BB, NN, 0, stream>>>(sc);
  const int q = BB * NN * NN / 4;
  outer_kernel<<<q / 256, 256, 0, stream>>>(sc, (float4*)out);
}